// Weights_31490700215135
// MI455X (gfx1250) — compile-verified
//
#include <hip/hip_runtime.h>
#include <hip/hip_bf16.h>

typedef __attribute__((ext_vector_type(2))) float v2f;
typedef __attribute__((ext_vector_type(8))) float v8f;

#define D_DIM 128

// ---------------------------------------------------------------------------
// Pass 0: zero the per-segment partition accumulator (400 KB, L2-resident).
// ---------------------------------------------------------------------------
__global__ void zero_f32(float* __restrict__ p, int n) {
    int i = blockIdx.x * blockDim.x + threadIdx.x;
    if (i < n) p[i] = 0.0f;
}

// ---------------------------------------------------------------------------
// Pass 1: one wave per 16 rows.
//   D = A x B with A = w broadcast (16x4 chunk), B = features^T (4x16 chunk),
//   accumulated over 32 chunks via V_WMMA_F32_16X16X4_F32 (fp32-exact mat-vec).
//   Every lane ends with dot(features[row_base + lane%16], w) in c[0].
//   Lanes 0-15 then exp(), store logit to d_out, and atomically add the logit
//   into partition[phrase_id[row]].
// ---------------------------------------------------------------------------
__global__ __launch_bounds__(256) void logit_partition_kernel(
        const float* __restrict__ feat,
        const float* __restrict__ w,
        const int*   __restrict__ pid,
        float*       __restrict__ logit_out,
        float*       __restrict__ part,
        int N) {
    const int lane = threadIdx.x & 31;
    const int wave = blockIdx.x * (blockDim.x >> 5) + (threadIdx.x >> 5);
    const long long row_base = (long long)wave * 16;
    if (row_base >= N) return;

    const int hi = lane >> 4;        // 0: lanes 0-15 (K=0,1), 1: lanes 16-31 (K=2,3)
    const int n  = lane & 15;        // which row of the 16-row tile this lane feeds
    long long row = row_base + n;
    if (row > (long long)N - 1) row = (long long)N - 1;   // clamp tail tile

    const float* frow = feat + row * D_DIM + 2 * hi;      // B operand stream
    const float* wcol = w + 2 * hi;                       // A operand stream

    float dotv;
#if __has_builtin(__builtin_amdgcn_wmma_f32_16x16x4_f32)
    v8f c = {};
#pragma unroll 8
    for (int j = 0; j < D_DIM / 4; ++j) {
        v2f b = *(const v2f*)(frow + 4 * j);   // features[row][4j+2hi .. +1]
        v2f a = *(const v2f*)(wcol + 4 * j);   // w[4j+2hi .. +1]
        // 8 args: (neg_a, A, neg_b, B, c_mod, C, reuse_a, reuse_b)
        c = __builtin_amdgcn_wmma_f32_16x16x4_f32(false, a, false, b,
                                                  (short)0, c, false, false);
    }
    dotv = c[0];   // independent of M: full dot product for row (row_base + n)
#else
    // VALU fallback (same data assignment): each lane covers half the columns,
    // partner lane (xor 16) covers the other half.
    float acc = 0.0f;
#pragma unroll 8
    for (int j = 0; j < D_DIM / 4; ++j) {
        acc += frow[4 * j] * wcol[4 * j] + frow[4 * j + 1] * wcol[4 * j + 1];
    }
    dotv = acc + __shfl_xor(acc, 16, 32);
#endif

    if (hi == 0) {
        long long r = row_base + n;
        if (r < N) {
            float lg = expf(dotv);
            logit_out[r] = lg;                       // stash logit in d_out
            atomicAdd(&part[pid[r]], lg);            // global_atomic_add_f32
        }
    }
}

// ---------------------------------------------------------------------------
// Pass 2: weights = logit / partition[phrase_id]  (in-place on d_out).
// ---------------------------------------------------------------------------
__global__ void normalize_kernel(float* __restrict__ out,
                                 const float* __restrict__ part,
                                 const int* __restrict__ pid, int N) {
    int i = blockIdx.x * blockDim.x + threadIdx.x;
    if (i < N) out[i] = out[i] / part[pid[i]];
}

extern "C" void kernel_launch(void* const* d_in, const int* in_sizes, int n_in,
                              void* d_out, int out_size, void* d_ws, size_t ws_size,
                              hipStream_t stream) {
    const float* feat = (const float*)d_in[0];   // [N, 128] fp32
    const float* w    = (const float*)d_in[1];   // [128] fp32
    const int*   pid  = (const int*)d_in[2];     // [N] int32
    const int N = in_sizes[2];
    const int S = in_sizes[3];

    float* out  = (float*)d_out;   // [N] fp32 (logits, then weights)
    float* part = (float*)d_ws;    // [S] fp32 partition accumulator

    zero_f32<<<(S + 255) / 256, 256, 0, stream>>>(part, S);

    const int tiles = (N + 15) / 16;       // one wave32 per 16-row tile
    const int wavesPerBlock = 256 / 32;    // 8 waves / block
    const int blocks = (tiles + wavesPerBlock - 1) / wavesPerBlock;
    logit_partition_kernel<<<blocks, 256, 0, stream>>>(feat, w, pid, out, part, N);

    normalize_kernel<<<(N + 255) / 256, 256, 0, stream>>>(out, part, pid, N);
}